// RSNALoss_48232482734267
// MI455X (gfx1250) — compile-verified
//
#include <hip/hip_runtime.h>

typedef float v2f __attribute__((ext_vector_type(2)));
typedef float v8f __attribute__((ext_vector_type(8)));

#define NCH   20            // 0..8: pred ch1..9 | 9..17: label ch1..9 | 18: label ch0 | 19: img bce
#define SLOT  33            // padded per-lane LDS stride in dwords (32 channels + 1 pad)
#define BLOCK 256
#define RPT   4             // rows per thread
#define LSEQ  8192
#define NBAT  128
#define CHUNK_ROWS (BLOCK * RPT)
#define NCHUNKS    (LSEQ / CHUNK_ROWS)   // 8

__global__ void rsna_zero_kernel(float* __restrict__ p, int n) {
    int i = blockIdx.x * blockDim.x + threadIdx.x;
    if (i < n) p[i] = 0.0f;
}

__global__ __launch_bounds__(BLOCK)
void rsna_main_kernel(const float* __restrict__ pred,
                      const float* __restrict__ label,
                      const int*   __restrict__ seq_lens,
                      float*       __restrict__ gacc) {
    __shared__ float lred[BLOCK / 32][32 * SLOT];   // per-wave staging (padded)
    __shared__ float blockAcc[NCH];

    const int b     = blockIdx.y;
    const int chunk = blockIdx.x;
    const int len   = seq_lens[b];
    const int l0    = chunk * CHUNK_ROWS;
    if (l0 >= len) return;          // uniform: whole block past the mask

    const int t = threadIdx.x;
    if (t < NCH) blockAcc[t] = 0.0f;

    float acc[NCH];
#pragma unroll
    for (int j = 0; j < NCH; ++j) acc[j] = 0.0f;

    const size_t rowbase = (size_t)b * LSEQ;
#pragma unroll
    for (int r = 0; r < RPT; ++r) {
        const int l = l0 + r * BLOCK + t;
        if (l < len) {
            const v2f* p2 = (const v2f*)(pred  + (rowbase + (size_t)l) * 10);  // 8B aligned
            const v2f* y2 = (const v2f*)(label + (rowbase + (size_t)l) * 10);
            float p[10], y[10];
#pragma unroll
            for (int q = 0; q < 5; ++q) {
                v2f a = p2[q]; v2f c = y2[q];
                p[2*q] = a.x; p[2*q+1] = a.y;
                y[2*q] = c.x; y[2*q+1] = c.y;
            }
#pragma unroll
            for (int c = 1; c < 10; ++c) acc[c - 1] += p[c];
#pragma unroll
            for (int c = 1; c < 10; ++c) acc[8 + c] += y[c];
            acc[18] += y[0];
            acc[19] += -(y[0] * __logf(p[0]) + (1.0f - y[0]) * __logf(1.0f - p[0]));
        }
    }

    // Stage per-lane partials to LDS so channels lie across lanes for WMMA.
    const int wave = t >> 5;
    const int lane = t & 31;
    float* ws = &lred[wave][lane * SLOT];
#pragma unroll
    for (int j = 0; j < NCH; ++j) ws[j] = acc[j];
#pragma unroll
    for (int j = NCH; j < 32; ++j) ws[j] = 0.0f;
    __syncthreads();

    // WMMA ones-reduction: D[m][n] += sum_k B[k][n].
    // B[k][n] = partial of source-lane (4g+k), channel (cb + n). Exact in f32.
    const int h = lane >> 4;        // which K half this lane supplies
    const int n = lane & 15;        // column / channel within group
    v2f ones; ones.x = 1.0f; ones.y = 1.0f;
    v8f d0 = {};                    // channels 0..15
    v8f d1 = {};                    // channels 16..31 (16..19 valid)
    const float* wbase = &lred[wave][0];
#pragma unroll
    for (int g = 0; g < 8; ++g) {
        const int s0 = (g * 4 + 2 * h) * SLOT;
        v2f b0, b1;
        b0.x = wbase[s0 + n];
        b0.y = wbase[s0 + SLOT + n];
        b1.x = wbase[s0 + 16 + n];
        b1.y = wbase[s0 + SLOT + 16 + n];
        d0 = __builtin_amdgcn_wmma_f32_16x16x4_f32(false, ones, false, b0,
                                                   (short)0, d0, false, false);
        d1 = __builtin_amdgcn_wmma_f32_16x16x4_f32(false, ones, false, b1,
                                                   (short)0, d1, false, false);
    }
    // D VGPR0, lanes 0..15 hold M=0, N=lane (documented C/D layout).
    if (lane < 16) atomicAdd(&blockAcc[lane], d0[0]);
    if (lane < 4)  atomicAdd(&blockAcc[16 + lane], d1[0]);
    __syncthreads();

    if (t < NCH) atomicAdd(&gacc[b * NCH + t], blockAcc[t]);
}

__global__ __launch_bounds__(128)
void rsna_final_kernel(const float* __restrict__ gacc,
                       const int*   __restrict__ seq_lens,
                       float*       __restrict__ out) {
    __shared__ float sl[128];
    __shared__ float st[128];
    const float w[9] = {0.0736196319f, 0.09202453988f, 0.1042944785f, 0.1042944785f,
                        0.1877300613f, 0.06257668712f, 0.06257668712f, 0.2346625767f,
                        0.0782208589f};
    const int b = threadIdx.x;
    const float* a = gacc + b * NCH;
    const float flen = (float)seq_lens[b];

    float exam = 0.0f;
#pragma unroll
    for (int c = 0; c < 9; ++c) {
        float pm = a[c]     / flen;
        float ym = a[9 + c] / flen;
        exam += w[c] * (-(ym * logf(pm) + (1.0f - ym) * logf(1.0f - pm)));
    }
    const float y0m = a[18] / flen;
    const float iw  = 0.0736196319f * y0m;
    sl[b] = exam + iw * a[19];   // exam + image loss contribution
    st[b] = iw * flen;           // extra total-weight contribution
    __syncthreads();

    for (int s = 64; s > 0; s >>= 1) {
        if (b < s) { sl[b] += sl[b + s]; st[b] += st[b + s]; }
        __syncthreads();
    }
    if (b == 0) {
        float sumw = 0.0f;
        for (int c = 0; c < 9; ++c) sumw += w[c];
        out[0] = sl[0] / ((float)NBAT * sumw + st[0]);
    }
}

extern "C" void kernel_launch(void* const* d_in, const int* in_sizes, int n_in,
                              void* d_out, int out_size, void* d_ws, size_t ws_size,
                              hipStream_t stream) {
    const float* pred  = (const float*)d_in[0];
    const float* label = (const float*)d_in[1];
    const int*   seq   = (const int*)d_in[2];
    float* gacc = (float*)d_ws;

    const int nacc = NBAT * NCH;
    hipLaunchKernelGGL(rsna_zero_kernel, dim3((nacc + 255) / 256), dim3(256), 0, stream,
                       gacc, nacc);

    dim3 grid(NCHUNKS, NBAT);
    hipLaunchKernelGGL(rsna_main_kernel, grid, dim3(BLOCK), 0, stream,
                       pred, label, seq, gacc);

    hipLaunchKernelGGL(rsna_final_kernel, dim3(1), dim3(128), 0, stream,
                       gacc, seq, (float*)d_out);
}